// GNN_22789096472662
// MI455X (gfx1250) — compile-verified
//
#include <hip/hip_runtime.h>
#include <hip/hip_bf16.h>
#include <math.h>

typedef __attribute__((ext_vector_type(2))) float v2f;
typedef __attribute__((ext_vector_type(8))) float v8f;

// ---------------------------------------------------------------------------
// Degree / normalization
// ---------------------------------------------------------------------------
__global__ void k_init_deg(float* __restrict__ deg, int n) {
    int i = blockIdx.x * blockDim.x + threadIdx.x;
    if (i < n) deg[i] = 1.0f;  // self-loop contribution
}

__global__ void k_accum_deg(const int* __restrict__ dst, float* __restrict__ deg, int e) {
    int i = blockIdx.x * blockDim.x + threadIdx.x;
    if (i < e) atomicAdd(&deg[dst[i]], 1.0f);
}

__global__ void k_rsqrt_inplace(float* __restrict__ deg, int n) {
    int i = blockIdx.x * blockDim.x + threadIdx.x;
    if (i < n) deg[i] = rsqrtf(deg[i]);  // deg >= 1 always (self-loops)
}

// ---------------------------------------------------------------------------
// Dense transform Y[16-row tile] = (RELU?)X @ W via V_WMMA_F32_16X16X4_F32.
// One wave per 16x16 output tile; 4 WMMA ops sweep K=16.
// A layout (32-bit A 16x4): lanes 0-15 hold K=0,1 in v0,v1; lanes 16-31 K=2,3.
// B layout mirrors: lanes 0-15 = N with K=0,1; lanes 16-31 K=2,3.
// D layout (32-bit C/D 16x16): vgpr r, lanes 0-15 -> (M=r, N=lane),
// lanes 16-31 -> (M=r+8, N=lane-16).
// ---------------------------------------------------------------------------
template <bool RELU, int NOUT>
__global__ void k_linear_wmma(const float* __restrict__ X,
                              const float* __restrict__ W,  // [16 x NOUT]
                              float* __restrict__ Y,        // [N x NOUT]
                              int numTiles) {
    const int lane = threadIdx.x & 31;
    const int wave = threadIdx.x >> 5;
    const int tile = blockIdx.x * (blockDim.x >> 5) + wave;
    if (tile >= numTiles) return;  // wave-uniform: EXEC stays all-1s for WMMA
    const int half = lane >> 4;    // 0: lanes 0-15, 1: lanes 16-31
    const int lr   = lane & 15;
    const int row0 = tile << 4;

    v8f acc = {};
#pragma unroll
    for (int ks = 0; ks < 4; ++ks) {
        const int ka = ks * 4 + half * 2;  // K index of first of the 2 A elems
        const float* xr = X + (size_t)(row0 + lr) * 16 + ka;
        float a0 = xr[0];
        float a1 = xr[1];
        if (RELU) { a0 = fmaxf(a0, 0.0f); a1 = fmaxf(a1, 0.0f); }
        v2f a; a.x = a0; a.y = a1;
        v2f b;
        if (NOUT == 16) {
            b.x = W[(ka + 0) * 16 + lr];
            b.y = W[(ka + 1) * 16 + lr];
        } else {  // zero-padded narrow W (e.g. 16x2 -> 16x16)
            b.x = (lr < NOUT) ? W[(ka + 0) * NOUT + lr] : 0.0f;
            b.y = (lr < NOUT) ? W[(ka + 1) * NOUT + lr] : 0.0f;
        }
        acc = __builtin_amdgcn_wmma_f32_16x16x4_f32(
            /*neg_a=*/false, a, /*neg_b=*/false, b,
            /*c_mod=*/(short)0, acc, /*reuse_a=*/false, /*reuse_b=*/false);
    }

#pragma unroll
    for (int r = 0; r < 8; ++r) {
        const int row = row0 + r + half * 8;
        if (NOUT == 16) {
            Y[(size_t)row * 16 + lr] = acc[r];
        } else if (lr < NOUT) {
            Y[(size_t)row * NOUT + lr] = acc[r];
        }
    }
}

// ---------------------------------------------------------------------------
// h[i,:] = bias + dinv[i]^2 * hw[i,:]   (bias + self-loop message, fused init)
// ---------------------------------------------------------------------------
template <int F>
__global__ void k_init_h(const float* __restrict__ hw, const float* __restrict__ dinv,
                         const float* __restrict__ bias, float* __restrict__ h, int n) {
    int idx = blockIdx.x * blockDim.x + threadIdx.x;
    if (idx < n * F) {
        int i = idx / F;
        int f = idx - i * F;
        float di = dinv[i];
        h[idx] = bias[f] + di * di * hw[idx];
    }
}

// ---------------------------------------------------------------------------
// Edge scatter: h[dst,:] += dinv[src]*dinv[dst] * hw[src,:]
// float4 (b128) gathers from L2-resident hw; non-returning fp32 atomics.
// ---------------------------------------------------------------------------
__global__ void k_scatter16(const int* __restrict__ src, const int* __restrict__ dst,
                            const float* __restrict__ dinv, const float* __restrict__ hw,
                            float* __restrict__ h, int e) {
    int i = blockIdx.x * blockDim.x + threadIdx.x;
    if (i >= e) return;
    const int s = src[i];
    const int d = dst[i];
    const float norm = dinv[s] * dinv[d];
    const float4* hp = (const float4*)(hw + (size_t)s * 16);
    float* out = h + (size_t)d * 16;
#pragma unroll
    for (int j = 0; j < 4; ++j) {
        float4 v = hp[j];
        atomicAdd(out + j * 4 + 0, norm * v.x);
        atomicAdd(out + j * 4 + 1, norm * v.y);
        atomicAdd(out + j * 4 + 2, norm * v.z);
        atomicAdd(out + j * 4 + 3, norm * v.w);
    }
}

__global__ void k_scatter2(const int* __restrict__ src, const int* __restrict__ dst,
                           const float* __restrict__ dinv, const float* __restrict__ hw,
                           float* __restrict__ h, int e) {
    int i = blockIdx.x * blockDim.x + threadIdx.x;
    if (i >= e) return;
    const int s = src[i];
    const int d = dst[i];
    const float norm = dinv[s] * dinv[d];
    const float2 v = *(const float2*)(hw + (size_t)s * 2);
    atomicAdd(h + (size_t)d * 2 + 0, norm * v.x);
    atomicAdd(h + (size_t)d * 2 + 1, norm * v.y);
}

// ---------------------------------------------------------------------------
// log_softmax over 2 classes
// ---------------------------------------------------------------------------
__global__ void k_logsoftmax2(const float* __restrict__ h, float* __restrict__ out, int n) {
    int i = blockIdx.x * blockDim.x + threadIdx.x;
    if (i < n) {
        float a0 = h[2 * i + 0];
        float a1 = h[2 * i + 1];
        float m  = fmaxf(a0, a1);
        float lse = m + logf(expf(a0 - m) + expf(a1 - m));
        out[2 * i + 0] = a0 - lse;
        out[2 * i + 1] = a1 - lse;
    }
}

// ---------------------------------------------------------------------------
// Launch
// inputs: x[200000,16], W1[16,16], b1[16], W2[16,2], b2[2], edge_index[2,3.2M]
// output: log_softmax [200000, 2] f32
// ---------------------------------------------------------------------------
extern "C" void kernel_launch(void* const* d_in, const int* in_sizes, int n_in,
                              void* d_out, int out_size, void* d_ws, size_t ws_size,
                              hipStream_t stream) {
    const float* x  = (const float*)d_in[0];
    const float* W1 = (const float*)d_in[1];
    const float* b1 = (const float*)d_in[2];
    const float* W2 = (const float*)d_in[3];
    const float* b2 = (const float*)d_in[4];
    const int*   ei = (const int*)d_in[5];

    const int N = in_sizes[0] / 16;
    const int E = in_sizes[5] / 2;
    const int* src = ei;
    const int* dst = ei + E;

    // Workspace carve-up: 37*N floats (~29.6 MB)
    float* ws   = (float*)d_ws;
    float* dinv = ws;                 // [N]   deg -> dinv (in place)
    float* hw1  = dinv + (size_t)N;   // [N,16]
    float* h1   = hw1 + (size_t)N*16; // [N,16]
    float* hw2  = h1  + (size_t)N*16; // [N,2]
    float* acc2 = hw2 + (size_t)N*2;  // [N,2]

    const int B = 256;
    auto G = [](long n, int b) { return (int)((n + b - 1) / b); };

    // Degrees + symmetric normalization
    k_init_deg<<<G(N, B), B, 0, stream>>>(dinv, N);
    k_accum_deg<<<G(E, B), B, 0, stream>>>(dst, dinv, E);
    k_rsqrt_inplace<<<G(N, B), B, 0, stream>>>(dinv, N);

    const int numTiles = N / 16;          // 12500 (exact)
    const int wavesPerBlk = 4;            // 128 threads, WGP-friendly wave32
    // Layer 1: hw1 = x @ W1 (WMMA f32), h1 = b1 + selfloop, scatter edges
    k_linear_wmma<false, 16><<<G(numTiles, wavesPerBlk), wavesPerBlk * 32, 0, stream>>>(
        x, W1, hw1, numTiles);
    k_init_h<16><<<G((long)N * 16, B), B, 0, stream>>>(hw1, dinv, b1, h1, N);
    k_scatter16<<<G(E, B), B, 0, stream>>>(src, dst, dinv, hw1, h1, E);

    // Layer 2: hw2 = relu(h1) @ W2 (WMMA f32, W2 zero-padded), then scatter
    k_linear_wmma<true, 2><<<G(numTiles, wavesPerBlk), wavesPerBlk * 32, 0, stream>>>(
        h1, W2, hw2, numTiles);
    k_init_h<2><<<G((long)N * 2, B), B, 0, stream>>>(hw2, dinv, b2, acc2, N);
    k_scatter2<<<G(E, B), B, 0, stream>>>(src, dst, dinv, hw2, acc2, E);

    // log-softmax -> d_out
    k_logsoftmax2<<<G(N, B), B, 0, stream>>>(acc2, (float*)d_out, N);
}